// EIGNN_scale_w_iter_T_52733608461015
// MI455X (gfx1250) — compile-verified
//
#include <hip/hip_runtime.h>
#include <math.h>

// ---- problem constants (match reference) ----
#define MDIM    128
#define GAMMA_F 0.8f
#define EPS_FF  1e-12f
#define NUM_INNER 21   // 20 while-loop body iterations (MAX_ITER always hit) + final inner()

typedef __attribute__((ext_vector_type(2))) float v2f;
typedef __attribute__((ext_vector_type(8))) float v8f;

// ---------------------------------------------------------------------------
// Kernel 1: FF = F^T F  (128x128), and accumulate squared Frobenius norm.
// 16384 output elements, 64 blocks x 256 threads, one element per thread.
// ---------------------------------------------------------------------------
__global__ void eignn_ftf_kernel(const float* __restrict__ F,
                                 float* __restrict__ FF,
                                 float* __restrict__ normsq) {
    int idx = blockIdx.x * blockDim.x + threadIdx.x;   // 0..16383
    int i = idx >> 7;     // row of FF
    int j = idx & 127;    // col of FF
    float acc = 0.0f;
#pragma unroll 8
    for (int k = 0; k < MDIM; ++k)
        acc = fmaf(F[k * MDIM + i], F[k * MDIM + j], acc);
    FF[idx] = acc;

    __shared__ float red[256];
    red[threadIdx.x] = acc * acc;
    __syncthreads();
    for (int s = 128; s > 0; s >>= 1) {
        if (threadIdx.x < s) red[threadIdx.x] += red[threadIdx.x + s];
        __syncthreads();
    }
    if (threadIdx.x == 0) atomicAdd(normsq, red[0]);
}

// ---------------------------------------------------------------------------
// Kernel 2: gF = FF / (||FF||_F + eps), in place. (FF symmetric => gF^T == gF)
// ---------------------------------------------------------------------------
__global__ void eignn_scale_kernel(float* __restrict__ FF,
                                   const float* __restrict__ normsq) {
    int idx = blockIdx.x * blockDim.x + threadIdx.x;
    float inv = 1.0f / (sqrtf(*normsq) + EPS_FF);
    FF[idx] *= inv;
}

// ---------------------------------------------------------------------------
// Kernel 3: sparse propagation  Pout[dst] += w * Zin[src]   (scatter-add)
// One wave32 per edge: lane l handles the float4 chunk [4l..4l+3] of the
// 128-float row (coalesced 512B row read, 4 f32 atomics per lane into L2).
// ---------------------------------------------------------------------------
__global__ void eignn_prop_kernel(const float* __restrict__ Zin,
                                  float* __restrict__ Pout,
                                  const int* __restrict__ esrc,
                                  const int* __restrict__ edst,
                                  const float* __restrict__ ew,
                                  int E) {
    int wave = (int)((blockIdx.x * blockDim.x + threadIdx.x) >> 5);
    int lane = threadIdx.x & 31;
    if (wave >= E) return;               // whole-wave uniform exit
    int s = esrc[wave];
    int d = edst[wave];
    float w = ew[wave];
    const float4* zrow = (const float4*)(Zin + (size_t)s * MDIM);
    float4 v = zrow[lane];
    float* prow = Pout + (size_t)d * MDIM + lane * 4;
    atomicAdd(prow + 0, w * v.x);
    atomicAdd(prow + 1, w * v.y);
    atomicAdd(prow + 2, w * v.z);
    atomicAdd(prow + 3, w * v.w);
}

// ---------------------------------------------------------------------------
// Kernel 4: fused  Zout = GAMMA * (P @ B) + X   with B = gF^T = gF (128x128).
// Full-precision f32 WMMA: V_WMMA_F32_16X16X4_F32, one wave per 16x16 tile,
// K=128 swept in 32 steps of 4. EXEC is all-1s (N%16==0, no divergence).
//
// f32 16x4 A fragment layout (ISA 7.12.2): lanes 0-15 hold K={0,1} in the two
// VGPRs, lanes 16-31 hold K={2,3}.  B (4x16) mirrors C/D row striping:
// VGPR0 = rows K and K+2, VGPR1 = rows K+1 and K+3 across the two lane halves.
// C/D: VGPR i = output rows i (lanes 0-15) and i+8 (lanes 16-31).
// ---------------------------------------------------------------------------
__global__ void eignn_gemm_axpy_kernel(const float* __restrict__ P,   // [N,128]
                                       const float* __restrict__ B,   // gF [128,128]
                                       const float* __restrict__ X,   // [N,128]
                                       float* __restrict__ Zout,      // [N,128]
                                       int N) {
    const int tilesN = MDIM / 16;                       // 8 column tiles
    int wave  = blockIdx.x * (blockDim.x >> 5) + (threadIdx.x >> 5);
    int lane  = threadIdx.x & 31;
    int tileM = wave / tilesN;
    int tileN = wave % tilesN;
    if (tileM * 16 >= N) return;                        // uniform per wave

    int half = lane >> 4;                               // 0: lanes 0-15, 1: lanes 16-31
    int l16  = lane & 15;
    int arow = tileM * 16 + l16;                        // A row for this lane
    int col  = tileN * 16 + l16;                        // B/D column for this lane

    v8f c = {};
#pragma unroll
    for (int k = 0; k < MDIM; k += 4) {
        int ka = k + half * 2;                          // this half's K base
        v2f a, b;
        a.x = P[(size_t)arow * MDIM + ka];              // contiguous pair -> b64 load
        a.y = P[(size_t)arow * MDIM + ka + 1];
        b.x = B[(size_t)ka * MDIM + col];
        b.y = B[(size_t)(ka + 1) * MDIM + col];
        // 8 args: (neg_a, A, neg_b, B, c_mod, C, reuse_a, reuse_b)
        c = __builtin_amdgcn_wmma_f32_16x16x4_f32(
                /*neg_a=*/false, a, /*neg_b=*/false, b,
                /*c_mod=*/(short)0, c, /*reuse_a=*/false, /*reuse_b=*/false);
    }

#pragma unroll
    for (int i = 0; i < 8; ++i) {
        int r = tileM * 16 + half * 8 + i;
        size_t off = (size_t)r * MDIM + col;
        Zout[off] = GAMMA_F * c[i] + X[off];
    }
}

// ---------------------------------------------------------------------------
// Host-side launch sequence (graph-capturable: only async memsets + kernels).
// ---------------------------------------------------------------------------
extern "C" void kernel_launch(void* const* d_in, const int* in_sizes, int n_in,
                              void* d_out, int out_size, void* d_ws, size_t ws_size,
                              hipStream_t stream) {
    const float* X    = (const float*)d_in[0];   // [N,128]
    const float* F    = (const float*)d_in[1];   // [128,128]
    const int*   esrc = (const int*)d_in[2];     // [E]
    const int*   edst = (const int*)d_in[3];     // [E]
    const float* ew   = (const float*)d_in[4];   // [E]

    const int N = in_sizes[0] / MDIM;
    const int E = in_sizes[2];
    const size_t zbytes = (size_t)N * MDIM * sizeof(float);

    float* Z = (float*)d_out;                    // Z lives in d_out across iterations

    // workspace layout: gF (64KB) | normsq (256B slot) | Pa | Pb
    char*  ws     = (char*)d_ws;
    float* gF     = (float*)ws;
    float* normsq = (float*)(ws + 65536);
    float* Pa     = (float*)(ws + 65536 + 256);
    float* Pb     = Pa + (size_t)N * MDIM;

    // --- g(F) = F^T F / (||F^T F||_F + eps); symmetric so gF^T == gF ---
    hipMemsetAsync(normsq, 0, sizeof(float), stream);
    eignn_ftf_kernel<<<64, 256, 0, stream>>>(F, gF, normsq);
    eignn_scale_kernel<<<64, 256, 0, stream>>>(gF, normsq);

    // --- Z0 = 0 ---
    hipMemsetAsync(Z, 0, zbytes, stream);

    const int propBlocks = (int)(((size_t)E * 32 + 255) / 256);
    const int totalWaves = (N / 16) * (MDIM / 16);
    const int gemmBlocks = (totalWaves + 7) / 8;         // 8 waves per 256-thread block

    // --- 20 fixed-point iterations (MAX_ITER is always hit) + final inner() ---
    for (int it = 0; it < NUM_INNER; ++it) {
        hipMemsetAsync(Pa, 0, zbytes, stream);
        eignn_prop_kernel<<<propBlocks, 256, 0, stream>>>(Z,  Pa, esrc, edst, ew, E);
        hipMemsetAsync(Pb, 0, zbytes, stream);
        eignn_prop_kernel<<<propBlocks, 256, 0, stream>>>(Pa, Pb, esrc, edst, ew, E);
        eignn_gemm_axpy_kernel<<<gemmBlocks, 256, 0, stream>>>(Pb, gF, X, Z, N);
    }
}